// SSELayer_41308995452950
// MI455X (gfx1250) — compile-verified
//
#include <hip/hip_runtime.h>
#include <hip/hip_bf16.h>

// ---- problem constants (match reference) ----
#define NDIM   32
#define EDIM   16
#define HID    64
#define ZDIM   144          // HID + 2*NDIM + EDIM
#define ZPAD   148          // LDS row stride for z (bank-conflict free, 8B-aligned pairs)
#define YPAD   68           // LDS row stride for y

typedef float v2f __attribute__((ext_vector_type(2)));
typedef float v8f __attribute__((ext_vector_type(8)));

// ---------------------------------------------------------------------------
// Kernel 1: zero workspace accumulators (deg | acc_ef | acc_h), 81 floats/node
// ---------------------------------------------------------------------------
__global__ void sse_zero_ws(float* __restrict__ ws, size_t n) {
    size_t i = (size_t)blockIdx.x * blockDim.x + threadIdx.x;
    if (i < n) ws[i] = 0.0f;
}

// ---------------------------------------------------------------------------
// Kernel 2: edge scatter. One wave32 per edge.
//   acc_h[dst]  += h[src]   (64 f32, 2 per lane)
//   acc_ef[dst] += ef[e]    (16 f32, lanes 0..15)
//   deg[dst]    += 1        (lane 0)
// ---------------------------------------------------------------------------
__global__ __launch_bounds__(256) void sse_edge_scatter(
    const float* __restrict__ h, const float* __restrict__ ef,
    const int* __restrict__ src, const int* __restrict__ dst,
    float* __restrict__ deg, float* __restrict__ acc_ef,
    float* __restrict__ acc_h, int n_edges)
{
    int tid  = blockIdx.x * blockDim.x + threadIdx.x;
    int e    = tid >> 5;
    int lane = threadIdx.x & 31;
    if (e >= n_edges) return;

    int s = src[e];
    int d = dst[e];

    const float* hs = h     + (size_t)s * HID;
    float*       ah = acc_h + (size_t)d * HID;
    unsafeAtomicAdd(&ah[lane],      hs[lane]);
    unsafeAtomicAdd(&ah[lane + 32], hs[lane + 32]);

    if (lane < EDIM)
        unsafeAtomicAdd(&acc_ef[(size_t)d * EDIM + lane],
                        ef[(size_t)e * EDIM + lane]);
    if (lane == 0)
        unsafeAtomicAdd(&deg[d], 1.0f);
}

// ---------------------------------------------------------------------------
// Kernel 3: per-node MLP via V_WMMA_F32_16X16X4_F32.
// Block = 128 threads (4 waves) = 16 nodes. Wave j computes output cols
// [16j, 16j+16) of both layers.
//   z[16][144] = [ nf | deg*nf | sum_ef | sum_h ]   (LDS, stride ZPAD)
//   y = relu(z @ W1^T)   (LDS, stride YPAD)
//   out = y @ W2^T, predicated on deg>0
// Fragment addressing (ISA 7.12.2, 32-bit, wave32):
//   A 16x4 : lanes 0-15 -> M=lane,  K={k,k+1}; lanes 16-31 -> M=lane-16, K={k+2,k+3}
//   B 4x16 : lanes 0-15 -> N=lane,  K={k,k+1}; lanes 16-31 -> N=lane-16, K={k+2,k+3}
//   C/D    : VGPR r -> M = r + (lane>=16 ? 8 : 0), N = lane&15
// ---------------------------------------------------------------------------
__global__ __launch_bounds__(128) void sse_node_mlp(
    const float* __restrict__ h,  const float* __restrict__ nf,
    const float* __restrict__ W1, const float* __restrict__ W2,
    const float* __restrict__ deg, const float* __restrict__ acc_ef,
    const float* __restrict__ acc_h, float* __restrict__ out, int n_nodes)
{
    __shared__ float zt[16 * ZPAD];
    __shared__ float yt[16 * YPAD];
    __shared__ float degs[16];

    const int tid  = threadIdx.x;
    const int base = blockIdx.x * 16;

    if (tid < 16) degs[tid] = (base + tid < n_nodes) ? deg[base + tid] : 0.0f;

    // build z tile in LDS (2304 values, 18 per thread)
    for (int idx = tid; idx < 16 * ZDIM; idx += 128) {
        int m = idx / ZDIM;
        int c = idx - m * ZDIM;
        int node = base + m;
        float v = 0.0f;
        if (node < n_nodes) {
            if (c < NDIM)            v = nf[(size_t)node * NDIM + c];
            else if (c < 2 * NDIM)   v = deg[node] * nf[(size_t)node * NDIM + (c - NDIM)];
            else if (c < 2 * NDIM + EDIM)
                                     v = acc_ef[(size_t)node * EDIM + (c - 2 * NDIM)];
            else                     v = acc_h[(size_t)node * HID + (c - 2 * NDIM - EDIM)];
        }
        zt[m * ZPAD + c] = v;
    }
    __syncthreads();

    const int wave = tid >> 5;          // 0..3 : output column tile
    const int lane = tid & 31;
    const int m    = lane & 15;         // M (A row / D col) within tile
    const int koff = (lane < 16) ? 0 : 2;
    const int bro  = wave * 16 + m;     // output-feature row of W1/W2
    const int mrow = (lane < 16) ? 0 : 8;

    // ---- layer 1: z(16x144) @ W1^T -> y tile (16x16) ----
    v8f acc = {};
    for (int k = 0; k < ZDIM; k += 4) {
        v2f a, b;
        a[0] = zt[m * ZPAD + k + koff];
        a[1] = zt[m * ZPAD + k + koff + 1];
        b[0] = W1[(size_t)bro * ZDIM + k + koff];
        b[1] = W1[(size_t)bro * ZDIM + k + koff + 1];
        acc = __builtin_amdgcn_wmma_f32_16x16x4_f32(
            false, a, false, b, (short)0, acc, false, false);
    }
    #pragma unroll
    for (int r = 0; r < 8; ++r) {
        float v = acc[r];
        yt[(r + mrow) * YPAD + wave * 16 + m] = v > 0.0f ? v : 0.0f;  // ReLU
    }
    __syncthreads();

    // ---- layer 2: y(16x64) @ W2^T -> out tile (16x16) ----
    v8f acc2 = {};
    for (int k = 0; k < HID; k += 4) {
        v2f a, b;
        a[0] = yt[m * YPAD + k + koff];
        a[1] = yt[m * YPAD + k + koff + 1];
        b[0] = W2[(size_t)bro * HID + k + koff];
        b[1] = W2[(size_t)bro * HID + k + koff + 1];
        acc2 = __builtin_amdgcn_wmma_f32_16x16x4_f32(
            false, a, false, b, (short)0, acc2, false, false);
    }

    // ---- writeback: nodes with deg==0 keep old h ----
    #pragma unroll
    for (int r = 0; r < 8; ++r) {
        int row = base + r + mrow;
        if (row < n_nodes) {
            int col = wave * 16 + m;
            out[(size_t)row * HID + col] =
                (degs[r + mrow] > 0.0f) ? acc2[r] : h[(size_t)row * HID + col];
        }
    }
}

// ---------------------------------------------------------------------------
extern "C" void kernel_launch(void* const* d_in, const int* in_sizes, int n_in,
                              void* d_out, int out_size, void* d_ws, size_t ws_size,
                              hipStream_t stream) {
    const float* h   = (const float*)d_in[0];
    const float* nf  = (const float*)d_in[1];
    const float* ef  = (const float*)d_in[2];
    const int*   src = (const int*)d_in[3];
    const int*   dst = (const int*)d_in[4];
    const float* W1  = (const float*)d_in[5];
    const float* W2  = (const float*)d_in[6];
    float* out = (float*)d_out;

    const int n_nodes = in_sizes[1] / NDIM;   // nf is [N, 32]
    const int n_edges = in_sizes[3];          // src is [E]

    // workspace layout: deg[N] | acc_ef[N*16] | acc_h[N*64]
    float* deg    = (float*)d_ws;
    float* acc_ef = deg + n_nodes;
    float* acc_h  = acc_ef + (size_t)n_nodes * EDIM;

    const size_t zlen = (size_t)n_nodes * (1 + EDIM + HID);
    sse_zero_ws<<<(unsigned)((zlen + 255) / 256), 256, 0, stream>>>((float*)d_ws, zlen);

    const long long sthreads = (long long)n_edges * 32;
    sse_edge_scatter<<<(unsigned)((sthreads + 255) / 256), 256, 0, stream>>>(
        h, ef, src, dst, deg, acc_ef, acc_h, n_edges);

    sse_node_mlp<<<(unsigned)((n_nodes + 15) / 16), 128, 0, stream>>>(
        h, nf, W1, W2, deg, acc_ef, acc_h, out, n_nodes);
}